// DetectionPostProcessor_62414464745859
// MI455X (gfx1250) — compile-verified
//
#include <hip/hip_runtime.h>
#include <hip/hip_bf16.h>
#include <math.h>

#define BATCH   8
#define NIN     100000
#define KTOP    1000
#define NS      40
#define SPAD    48          // samples padded to 3 WMMA M-tiles
#define DET     300
#define NBINS   4096
#define CAP     4096
#define JT      63          // ceil(1000/16) column tiles
#define IB      4           // boxes per wave in the WMMA kernel (KTOP % IB == 0)
#define SCORE_TH 0.05f
#define NMS_TH   0.5f
#define GOLDEN   0.6180339887498949f

typedef float v2f __attribute__((ext_vector_type(2)));
typedef float v8f __attribute__((ext_vector_type(8)));

// ---------------- 1) score histogram ----------------
__global__ void k_hist(const float* __restrict__ scores, int* __restrict__ hist) {
    int g = blockIdx.x * blockDim.x + threadIdx.x;
    if (g >= BATCH * NIN) return;
    int b = g / NIN;
    float s = scores[g];
    if (s > SCORE_TH) {
        int bin = (int)(s * (float)NBINS);
        bin = bin < 0 ? 0 : (bin > NBINS - 1 ? NBINS - 1 : bin);
        atomicAdd(&hist[b * NBINS + bin], 1);
    }
}

// ---------------- 2) find threshold bin so suffix count >= KTOP ----------------
__global__ void k_thresh(const int* __restrict__ hist, int* __restrict__ tbin) {
    if (threadIdx.x != 0) return;
    int b = blockIdx.x;
    int acc = 0, tb = 0;
    for (int bin = NBINS - 1; bin >= 0; --bin) {
        acc += hist[b * NBINS + bin];
        if (acc >= KTOP) { tb = bin; break; }
    }
    tbin[b] = tb;
}

// ---------------- 3) compact candidate (score, idx) ----------------
__global__ void k_compact(const float* __restrict__ scores, const int* __restrict__ tbin,
                          int* __restrict__ ccount, int* __restrict__ cidx,
                          float* __restrict__ cscore) {
    int g = blockIdx.x * blockDim.x + threadIdx.x;
    if (g >= BATCH * NIN) return;
    int b = g / NIN, n = g % NIN;
    float s = scores[g];
    if (s > SCORE_TH) {
        int bin = (int)(s * (float)NBINS);
        bin = bin < 0 ? 0 : (bin > NBINS - 1 ? NBINS - 1 : bin);
        if (bin >= tbin[b]) {
            int pos = atomicAdd(&ccount[b], 1);
            if (pos < CAP) { cidx[b * CAP + pos] = n; cscore[b * CAP + pos] = s; }
        }
    }
}

// ---------------- 4) bitonic sort (desc) + gather + geometry + sample points ----------------
__global__ __launch_bounds__(1024) void k_sort_prep(
    const float* __restrict__ boxes, const int* __restrict__ labels,
    const int* __restrict__ ccount, const int* __restrict__ cidx,
    const float* __restrict__ cscore,
    float* __restrict__ top_score, float* __restrict__ top_box,
    int* __restrict__ top_label, float* __restrict__ geom, float* __restrict__ pts) {
    __shared__ float ls[CAP];
    __shared__ int   li[CAP];
    int b = blockIdx.x;
    int cnt = ccount[b]; if (cnt > CAP) cnt = CAP;
    for (int t = threadIdx.x; t < CAP; t += 1024) {
        if (t < cnt) { ls[t] = cscore[b * CAP + t]; li[t] = cidx[b * CAP + t]; }
        else         { ls[t] = -1e30f;              li[t] = 0x7fffffff; }
    }
    __syncthreads();
    for (int k = 2; k <= CAP; k <<= 1) {
        for (int j = k >> 1; j > 0; j >>= 1) {
            for (int t = threadIdx.x; t < CAP; t += 1024) {
                int p = t ^ j;
                if (p > t) {
                    bool desc = ((t & k) == 0);
                    float s0 = ls[t], s1 = ls[p];
                    int   i0 = li[t], i1 = li[p];
                    bool sw = desc ? (s0 < s1 || (s0 == s1 && i0 > i1))
                                   : (s0 > s1 || (s0 == s1 && i0 < i1));
                    if (sw) { ls[t] = s1; ls[p] = s0; li[t] = i1; li[p] = i0; }
                }
            }
            __syncthreads();
        }
    }
    // gather top-K, emit derived geometry
    for (int t = threadIdx.x; t < KTOP; t += 1024) {
        float sc = ls[t]; int idx = li[t];
        bool ok = (sc > SCORE_TH) && (idx < NIN);
        float cx = 0.f, cy = 0.f, w = 1.f, h = 1.f, ang = 0.f; int lab = -1;
        if (ok) {
            const float* bp = boxes + ((size_t)b * NIN + idx) * 5;
            cx = bp[0]; cy = bp[1]; w = bp[2]; h = bp[3]; ang = bp[4];
            lab = labels[(size_t)b * NIN + idx];
        }
        int o = b * KTOP + t;
        top_score[o] = ok ? sc : -1.0f;
        float* tb = top_box + (size_t)o * 5;
        tb[0] = ok ? cx : 0.f; tb[1] = ok ? cy : 0.f; tb[2] = ok ? w : 0.f;
        tb[3] = ok ? h : 0.f;  tb[4] = ok ? ang : 0.f;
        top_label[o] = lab;
        float c = cosf(ang), sn = sinf(ang);
        float* g8 = geom + (size_t)o * 8;
        g8[0] = c; g8[1] = sn;
        g8[2] = -(c * cx + sn * cy);     // tx for qx
        g8[3] =  (sn * cx - c * cy);     // ty for qy
        g8[4] = ok ? 0.5f * w : -1.0f;   // half-width (invalid -> nothing inside)
        g8[5] = ok ? 0.5f * h : -1.0f;
        g8[6] = w * h;                   // area
        g8[7] = (float)lab;              // label for same-class gate
    }
    __syncthreads();
    // deterministic sample points (padded rows 40..47 zeroed; masked in WMMA kernel)
    for (int r = threadIdx.x; r < KTOP * SPAD; r += 1024) {
        int i = r / SPAD, t = r % SPAD;
        float px = 0.f, py = 0.f;
        if (t < NS) {
            const float* tb = top_box + ((size_t)(b * KTOP + i)) * 5;
            const float* g8 = geom   + ((size_t)(b * KTOP + i)) * 8;
            float cx = tb[0], cy = tb[1], w = tb[2], h = tb[3];
            float c = g8[0], sn = g8[1];
            float sd = (float)t + 0.5f;
            float ux = sd / (float)NS - 0.5f;
            float uy = fmodf(sd * GOLDEN, 1.0f) - 0.5f;
            float lx = ux * w, ly = uy * h;
            px = cx + c * lx - sn * ly;
            py = cy + sn * lx + c * ly;
        }
        float* pp = pts + ((size_t)b * KTOP * SPAD + r) * 2;
        pp[0] = px; pp[1] = py;
    }
}

// ---------------- 5) WMMA containment-count kernel ----------------
// One wave per (IB-box block, 16-wide j tile): qx/qy tiles via V_WMMA_F32_16X16X4_F32.
// B operands (c,s,tx | -s,c,ty per column) loaded once per wave, reused for 24 WMMAs.
// A-operand loads are branchless: all lanes load, hi half cndmask'd to (1,0).
__global__ __launch_bounds__(256) void k_wmma_count(
    const float* __restrict__ pts, const float* __restrict__ geom,
    unsigned char* __restrict__ cnt) {
    const int NIB = KTOP / IB;                         // 250 box-blocks
    int wave = blockIdx.x * (blockDim.x >> 5) + (threadIdx.x >> 5);
    const int TASKS = BATCH * NIB * JT;
    if (wave >= TASKS) return;                         // uniform per wave -> EXEC all-1 for WMMA
    int b    = wave / (NIB * JT);
    int rem  = wave % (NIB * JT);
    int ib   = rem / JT;                               // box block: boxes ib*IB .. ib*IB+IB-1
    int jt   = rem % JT;

    int lane = threadIdx.x & 31;
    int n    = lane & 15;
    int j    = jt * 16 + n;
    bool jok = (j < KTOP);
    bool lo  = (lane < 16);

    const float* g8 = geom + ((size_t)(b * KTOP) + (jok ? j : 0)) * 8;
    float c = g8[0], sn = g8[1], tx = g8[2], ty = g8[3];
    float hw = jok ? g8[4] : -1.0f;
    float hh = jok ? g8[5] : -1.0f;

    // B operand VGPR layout: lanes 0-15 hold K=0,1; lanes 16-31 hold K=2,3 (N = lane%16)
    v2f bx, by;
    bx.x = lo ?  c  : tx;  bx.y = lo ? sn : 0.0f;
    by.x = lo ? -sn : ty;  by.y = lo ? c  : 0.0f;

    v8f z = {};
    int i0 = ib * IB;
    const float* pbase = pts + ((size_t)b * KTOP + i0) * (SPAD * 2);

#pragma unroll
    for (int ii = 0; ii < IB; ++ii) {
        const float* prow = pbase + ii * (SPAD * 2);
        int count = 0;
#pragma unroll
        for (int sub = 0; sub < 3; ++sub) {
            // Branchless A operand: every lane loads row (lane&15); hi lanes select (1,0).
            v2f av = *(const v2f*)(prow + (sub * 16 + n) * 2);
            v2f a;
            a.x = lo ? av.x : 1.0f;
            a.y = lo ? av.y : 0.0f;
            v8f qx = __builtin_amdgcn_wmma_f32_16x16x4_f32(false, a, false, bx, (short)0, z, false, false);
            v8f qy = __builtin_amdgcn_wmma_f32_16x16x4_f32(false, a, false, by, (short)0, z, false, false);
#pragma unroll
            for (int r = 0; r < 8; ++r) {
                int srow = sub * 16 + r + (lo ? 0 : 8);   // C/D layout: VGPR r = M=r (lo) / M=r+8 (hi)
                bool inside = (srow < NS) && (fabsf(qx[r]) <= hw) && (fabsf(qy[r]) <= hh);
                count += inside ? 1 : 0;
            }
        }
        count += __shfl_xor(count, 16);                   // fold rows 8-15 half onto rows 0-7 half
        if (lo && jok)
            cnt[((size_t)(b * KTOP) + (i0 + ii)) * KTOP + j] = (unsigned char)count;
    }
}

// ---------------- 6) IoU + suppression bitmask (wave32 ballot) ----------------
__global__ void k_supp(const unsigned char* __restrict__ cnt, const float* __restrict__ geom,
                       unsigned int* __restrict__ supp) {
    int g = blockIdx.x * blockDim.x + threadIdx.x;   // exactly B*KTOP*32*32 threads
    int lane = g & 31;
    int rest = g >> 5;
    int w = rest & 31; rest >>= 5;
    int i = rest % KTOP;
    int b = rest / KTOP;
    int k = w * 32 + lane;
    bool pred = false;
    if (k < KTOP) {
        const float* gi = geom + ((size_t)(b * KTOP) + i) * 8;
        const float* gk = geom + ((size_t)(b * KTOP) + k) * 8;
        float ai = gi[6], ak = gk[6];
        float fik = (float)cnt[((size_t)(b * KTOP) + i) * KTOP + k] * (1.0f / NS);
        float fki = (float)cnt[((size_t)(b * KTOP) + k) * KTOP + i] * (1.0f / NS);
        float inter = 0.5f * (fki * ak + fik * ai);
        float iou = inter / (ak + ai - inter + 1e-7f);
        pred = (gi[7] == gk[7]) && (iou > NMS_TH);   // label gate == class-offset trick
    }
    unsigned long long m = __ballot(pred);
    if (lane == 0) supp[((size_t)(b * KTOP) + i) * 32 + w] = (unsigned int)m;
}

// ---------------- 7) sequential greedy NMS: one wave per image ----------------
__global__ __launch_bounds__(32) void k_nms(const unsigned int* __restrict__ supp,
                                            const float* __restrict__ top_score,
                                            unsigned int* __restrict__ keepw) {
    int b = blockIdx.x;
    int lane = threadIdx.x;            // lane owns keep word `lane` (32 words = 1024 bits)
    unsigned int kw = 0u;
    for (int i = 0; i < KTOP; ++i) {
        unsigned int srow = supp[((size_t)(b * KTOP) + i) * 32 + lane];
        int sup = __any((kw & srow) != 0u);
        if (!sup && top_score[b * KTOP + i] > SCORE_TH && (i >> 5) == lane)
            kw |= 1u << (i & 31);
    }
    keepw[b * 32 + lane] = kw;
}

// ---------------- 8) compact kept detections to output ----------------
__global__ __launch_bounds__(256) void k_output(
    const unsigned int* __restrict__ keepw, const float* __restrict__ top_score,
    const float* __restrict__ top_box, const int* __restrict__ top_label,
    float* __restrict__ out_boxes, float* __restrict__ out_scores,
    float* __restrict__ out_labels) {
    __shared__ int order[DET];
    __shared__ int sh_nk;
    int b = blockIdx.x;
    if (threadIdx.x == 0) {
        int nk = 0;
        for (int i = 0; i < KTOP; ++i) {
            if ((keepw[b * 32 + (i >> 5)] >> (i & 31)) & 1u) {
                if (nk < DET) order[nk] = i;
                nk++;
            }
        }
        sh_nk = nk;
    }
    __syncthreads();
    int nk = sh_nk; if (nk > DET) nk = DET;
    for (int d = threadIdx.x; d < DET; d += blockDim.x) {
        bool sel = d < nk;
        int i = sel ? order[d] : 0;
        const float* tb = top_box + ((size_t)(b * KTOP) + i) * 5;
        float* ob = out_boxes + ((size_t)(b * DET) + d) * 5;
#pragma unroll
        for (int f = 0; f < 5; ++f) ob[f] = sel ? tb[f] : 0.0f;
        out_scores[b * DET + d] = sel ? top_score[b * KTOP + i] : 0.0f;
        out_labels[b * DET + d] = sel ? (float)top_label[b * KTOP + i] : -1.0f;
    }
}

extern "C" void kernel_launch(void* const* d_in, const int* in_sizes, int n_in,
                              void* d_out, int out_size, void* d_ws, size_t ws_size,
                              hipStream_t stream) {
    const float* boxes  = (const float*)d_in[0];
    const float* scores = (const float*)d_in[1];
    const int*   labels = (const int*)d_in[2];

    char* ws = (char*)d_ws;
    size_t off = 0;
    auto alloc = [&](size_t bytes) { size_t o = off; off = (off + bytes + 255) & ~(size_t)255; return o; };

    // zero-initialized region first (hist + candidate counters)
    size_t o_hist  = alloc((size_t)BATCH * NBINS * 4);
    size_t o_ccnt  = alloc((size_t)BATCH * 4);
    size_t zero_sz = off;
    size_t o_tbin  = alloc((size_t)BATCH * 4);
    size_t o_cidx  = alloc((size_t)BATCH * CAP * 4);
    size_t o_cscr  = alloc((size_t)BATCH * CAP * 4);
    size_t o_tsc   = alloc((size_t)BATCH * KTOP * 4);
    size_t o_tbox  = alloc((size_t)BATCH * KTOP * 5 * 4);
    size_t o_tlab  = alloc((size_t)BATCH * KTOP * 4);
    size_t o_geom  = alloc((size_t)BATCH * KTOP * 8 * 4);
    size_t o_pts   = alloc((size_t)BATCH * KTOP * SPAD * 2 * 4);
    size_t o_cnt   = alloc((size_t)BATCH * KTOP * KTOP);
    size_t o_supp  = alloc((size_t)BATCH * KTOP * 32 * 4);
    size_t o_keep  = alloc((size_t)BATCH * 32 * 4);
    (void)ws_size;

    int*            hist   = (int*)(ws + o_hist);
    int*            ccnt   = (int*)(ws + o_ccnt);
    int*            tbin   = (int*)(ws + o_tbin);
    int*            cidx   = (int*)(ws + o_cidx);
    float*          cscr   = (float*)(ws + o_cscr);
    float*          tsc    = (float*)(ws + o_tsc);
    float*          tbox   = (float*)(ws + o_tbox);
    int*            tlab   = (int*)(ws + o_tlab);
    float*          geom   = (float*)(ws + o_geom);
    float*          pts    = (float*)(ws + o_pts);
    unsigned char*  cntm   = (unsigned char*)(ws + o_cnt);
    unsigned int*   supp   = (unsigned int*)(ws + o_supp);
    unsigned int*   keepw  = (unsigned int*)(ws + o_keep);

    float* out_boxes  = (float*)d_out;                         // [B,300,5]
    float* out_scores = out_boxes + (size_t)BATCH * DET * 5;   // [B,300]
    float* out_labels = out_scores + (size_t)BATCH * DET;      // [B,300]

    hipMemsetAsync(ws, 0, zero_sz, stream);

    k_hist   <<<(BATCH * NIN + 255) / 256, 256, 0, stream>>>(scores, hist);
    k_thresh <<<BATCH, 32, 0, stream>>>(hist, tbin);
    k_compact<<<(BATCH * NIN + 255) / 256, 256, 0, stream>>>(scores, tbin, ccnt, cidx, cscr);
    k_sort_prep<<<BATCH, 1024, 0, stream>>>(boxes, labels, ccnt, cidx, cscr,
                                            tsc, tbox, tlab, geom, pts);
    {
        const int NIB = KTOP / IB;
        const int TASKS = BATCH * NIB * JT;               // waves
        k_wmma_count<<<(TASKS + 7) / 8, 256, 0, stream>>>(pts, geom, cntm);
    }
    k_supp  <<<(BATCH * KTOP * 32 * 32) / 256, 256, 0, stream>>>(cntm, geom, supp);
    k_nms   <<<BATCH, 32, 0, stream>>>(supp, tsc, keepw);
    k_output<<<BATCH, 256, 0, stream>>>(keepw, tsc, tbox, tlab,
                                        out_boxes, out_scores, out_labels);
}